// ChamferLossV2_652835029200
// MI455X (gfx1250) — compile-verified
//
#include <hip/hip_runtime.h>
#include <math.h>

typedef __attribute__((ext_vector_type(2))) float v2f;
typedef __attribute__((ext_vector_type(8))) float v8f;

#define BIGF  1e10f
#define CHUNK 2048          // Y columns staged per pass: 16B/col -> 32KB LDS

// One wave owns 16 rows of X; block stages Y columns into LDS as
// (x, y, z, |t|^2 + maskbias), then each wave sweeps the chunk in 16-column
// tiles (2 tiles per iteration, loads software-pipelined): ds_load_b128 +
// v_cndmask builds the WMMA B operand, V_WMMA_F32_16X16X4_F32 gives the
// 16x16 cross terms, packed VALU (v_pk_fma / v_maxmin) assembles clamped
// squared distances and keeps per-row running mins. Rows are exclusively
// owned -> plain stores, no init, no atomics.
__global__ void chamfer_rowmin_wmma(const float* __restrict__ X,
                                    const float* __restrict__ Y,
                                    const int*   __restrict__ labelY,
                                    float*       __restrict__ rowmin,
                                    int nx, int ny, int tilesPerBatch)
{
    __shared__ float4 sY[CHUNK];

    const int wave = threadIdx.x >> 5;       // 8 waves / block -> 128 rows
    const int lane = threadIdx.x & 31;
    const int half = lane >> 4;              // 0: lanes 0-15, 1: lanes 16-31
    const int idx  = lane & 15;

    const int b    = blockIdx.x / tilesPerBatch;
    const int tile = blockIdx.x % tilesPerBatch;
    const int rb   = tile * 128 + wave * 16; // first row owned by this wave

    const float* Xb = X + (size_t)b * nx * 3;
    const float* Yb = Y + (size_t)b * ny * 3;
    const int*   Lb = labelY ? (labelY + (size_t)b * ny) : nullptr;

    // A matrix 16x4 (row = rb+idx, K = {x,y,z,0}); branch-free select.
    const float* xa = Xb + 3 * (rb + idx);
    const float ax = xa[0], ay = xa[1], az = xa[2];
    v2f a;
    a[0] = half ? az   : ax;
    a[1] = half ? 0.0f : ay;

    // |p|^2 for the 8 rows covered by this half's D registers:
    // D VGPR r -> row rb + 8*half + r
    const int rsel = rb + 8 * half;
    float pn8[8];
#pragma unroll
    for (int r = 0; r < 8; ++r) {
        const float* xp = Xb + 3 * (rsel + r);
        pn8[r] = xp[0]*xp[0] + xp[1]*xp[1] + xp[2]*xp[2];
    }

    float rmin[8];
#pragma unroll
    for (int r = 0; r < 8; ++r) rmin[r] = 4.0f * BIGF;

    // process one 16-column tile held in t = (x,y,z,|t|^2+bias)
    auto process = [&](const float4 t) {
        v2f bm;
        bm[0] = half ? t.z  : t.x;
        bm[1] = half ? 0.0f : t.y;
        v8f c = {};
        c = __builtin_amdgcn_wmma_f32_16x16x4_f32(false, a, false, bm,
                                                  (short)0, c, false, false);
#pragma unroll
        for (int r = 0; r < 8; ++r) {
            float d = fmaxf(pn8[r] + t.w - 2.0f * c[r], 0.0f);
            rmin[r] = fminf(rmin[r], d);
        }
    };

    for (int base = 0; base < ny; base += CHUNK) {
        const int cols = min(CHUNK, ny - base);

        // Cooperative staging: (x, y, z, |t|^2 [+BIG if masked out]).
        for (int j = threadIdx.x; j < cols; j += 256) {
            const float* yp = Yb + 3 * (base + j);
            const float x = yp[0], y = yp[1], z = yp[2];
            float tn = x*x + y*y + z*z;
            if (Lb) tn += (Lb[base + j] == 1) ? 0.0f : BIGF;
            sY[j] = make_float4(x, y, z, tn);
        }
        __syncthreads();

        // 2 tiles per iteration: both ds_load_b128 issue before the first
        // tile's compute, hiding LDS latency behind WMMA + VALU.
        int cb = 0;
        for (; cb + 32 <= cols; cb += 32) {
            const float4 t0 = sY[cb + idx];
            const float4 t1 = sY[cb + 16 + idx];
            process(t0);
            process(t1);
        }
        for (; cb < cols; cb += 16)          // tail (cols % 32 == 16)
            process(sY[cb + idx]);

        __syncthreads();   // before next chunk overwrites sY
    }

    // min-reduce each rmin[r] across the 16 lanes of this half-wave
#pragma unroll
    for (int r = 0; r < 8; ++r) {
        float v = rmin[r];
        v = fminf(v, __shfl_xor(v, 1, 32));
        v = fminf(v, __shfl_xor(v, 2, 32));
        v = fminf(v, __shfl_xor(v, 4, 32));
        v = fminf(v, __shfl_xor(v, 8, 32));
        rmin[r] = v;
    }
    float* out = rowmin + (size_t)b * nx + rsel;
    if (idx == 0) {        // lane 0 -> rows rb..rb+7, lane 16 -> rb+8..rb+15
#pragma unroll
        for (int r = 0; r < 8; ++r) out[r] = rmin[r];
    }
}

// Final masked means + combine: one block, 256 threads.
__global__ void chamfer_finalize(const float* __restrict__ rowminA, // [B*N] pred->target (sq)
                                 const float* __restrict__ rowminB, // [B*M] target->pred (sq)
                                 const int*   __restrict__ label,   // [B*N]
                                 float*       __restrict__ out,
                                 int nb, int nN, int nM)
{
    __shared__ float s1[256], s2[256], sc[256];
    const int tid = threadIdx.x;
    float total = 0.0f;
    for (int b = 0; b < nb; ++b) {
        float sum1 = 0.0f, cnt = 0.0f, sum2 = 0.0f;
        for (int n = tid; n < nN; n += 256) {
            if (label[(size_t)b * nN + n] == 1) {
                sum1 += sqrtf(rowminA[(size_t)b * nN + n]);
                cnt  += 1.0f;
            }
        }
        for (int m = tid; m < nM; m += 256)
            sum2 += sqrtf(rowminB[(size_t)b * nM + m]);
        s1[tid] = sum1; s2[tid] = sum2; sc[tid] = cnt;
        __syncthreads();
        for (int s = 128; s > 0; s >>= 1) {
            if (tid < s) {
                s1[tid] += s1[tid + s];
                s2[tid] += s2[tid + s];
                sc[tid] += sc[tid + s];
            }
            __syncthreads();
        }
        if (tid == 0) {
            float c = fmaxf(sc[0], 1.0f);
            total += 0.5f * (s1[0] / c + s2[0] / (float)nM);
        }
        __syncthreads();
    }
    if (tid == 0) out[0] = total / (float)nb;   // LOSS_WEIGHT == 1.0
}

extern "C" void kernel_launch(void* const* d_in, const int* in_sizes, int n_in,
                              void* d_out, int out_size, void* d_ws, size_t ws_size,
                              hipStream_t stream) {
    const float* pred  = (const float*)d_in[0];   // [B*N,3]
    const float* targ  = (const float*)d_in[1];   // [B*M,3]
    const int*   label = (const int*)  d_in[2];   // [B*N]

    const int nb = in_sizes[3];                   // B (nums has B entries)
    const int nN = in_sizes[2] / nb;              // N from label size
    const int nM = in_sizes[1] / (3 * nb);        // M from target size

    float* rowminA = (float*)d_ws;                // B*N floats
    float* rowminB = rowminA + (size_t)nb * nN;   // B*M floats

    const int tilesA = nN / 128;                  // rows per block = 128
    const int tilesB = nM / 128;

    // pred -> target: rows = pred, cols = target, no mask bias
    chamfer_rowmin_wmma<<<nb * tilesA, 256, 0, stream>>>(
        pred, targ, nullptr, rowminA, nN, nM, tilesA);
    // target -> pred: rows = target, cols = pred, +BIG bias on masked preds
    chamfer_rowmin_wmma<<<nb * tilesB, 256, 0, stream>>>(
        targ, pred, label, rowminB, nM, nN, tilesB);

    chamfer_finalize<<<1, 256, 0, stream>>>(
        rowminA, rowminB, label, (float*)d_out, nb, nN, nM);
}